// SubDualNet_90142773608473
// MI455X (gfx1250) — compile-verified
//
#include <hip/hip_runtime.h>

// SubDualNet fused pipeline for MI455X (gfx1250, wave32):
//   X   = (primal @ W2^T + b2) - (last_primal @ W3^T + b3)        [kernel 1, WMMA f32]
//   out = dual @ W1^T + b1 - sigma*rhs                            [kernel 2, WMMA f32]
//   out[rows[e]] += sigma * vals[e] * X[cols[e]]   (f32 atomics)  [kernel 3]
//   out = leaky_relu(out, 0.01)                                   [kernel 4]
// X lives in d_ws (200000*128*4 = 102.4 MB). X + out = 153 MB < 192 MB L2,
// so the sparse gather/scatter traffic stays on-chip.

#define HID 128
#define LDSS 132            // padded LDS row stride (bank-rotation, 16B aligned)
#define NEG_SLOPE 0.01f

typedef float v2f __attribute__((ext_vector_type(2)));
typedef float v8f __attribute__((ext_vector_type(8)));

__device__ __forceinline__ v8f wmma_f32(v2f a, v2f b, v8f c) {
  // V_WMMA_F32_16X16X4_F32: D = A(16x4) x B(4x16) + C(16x16), fp32 tensor path
  return __builtin_amdgcn_wmma_f32_16x16x4_f32(
      /*neg_a=*/false, a, /*neg_b=*/false, b,
      /*c_mod=*/(short)0, c, /*reuse_a=*/false, /*reuse_b=*/false);
}

// ---------------------------------------------------------------------------
// Kernel 1: X = (primal @ W2^T + b2) - (last_primal @ W3^T + b3)
// 1 block = 16 rows of X, 128 threads = 4 waves, each wave owns 2 N-tiles.
// ---------------------------------------------------------------------------
__global__ __launch_bounds__(128)
void fused_x_kernel(const float* __restrict__ primal,
                    const float* __restrict__ last_primal,
                    const float* __restrict__ W2, const float* __restrict__ b2,
                    const float* __restrict__ W3, const float* __restrict__ b3,
                    float* __restrict__ X) {
  __shared__ float Ap[16 * LDSS];
  __shared__ float Al[16 * LDSS];
  const int m0  = blockIdx.x * 16;
  const int tid = threadIdx.x;

  // cooperative coalesced load of two 16x128 fp32 tiles into padded LDS
  const float4* gp = (const float4*)(primal      + (size_t)m0 * HID);
  const float4* gl = (const float4*)(last_primal + (size_t)m0 * HID);
#pragma unroll
  for (int j = 0; j < 4; ++j) {
    const int f   = tid + j * 128;   // float4 index, 512 total
    const int row = f >> 5;          // 32 float4 per row
    const int c4  = f & 31;
    *(float4*)&Ap[row * LDSS + c4 * 4] = gp[f];
    *(float4*)&Al[row * LDSS + c4 * 4] = gl[f];
  }
  __syncthreads();

  const int lane = tid & 31;
  const int wave = tid >> 5;
  const int r    = lane & 15;            // A-frag row for this lane
  const int koff = (lane >> 4) << 1;     // 0 for lanes 0-15, 2 for lanes 16-31

#pragma unroll
  for (int t = 0; t < 2; ++t) {
    const int n0 = (wave * 2 + t) * 16;
    const int n  = n0 + (lane & 15);     // B-frag / output column
    v8f acc  = {};                       // primal      @ W2^T
    v8f acc2 = {};                       // last_primal @ W3^T
#pragma unroll
    for (int k0 = 0; k0 < HID; k0 += 4) {
      const int kk = k0 + koff;
      v2f a, b, a2, b2v;
      a.x   = Ap[r * LDSS + kk];  a.y   = Ap[r * LDSS + kk + 1];
      b.x   = W2[n * HID + kk];   b.y   = W2[n * HID + kk + 1];
      acc   = wmma_f32(a, b, acc);
      a2.x  = Al[r * LDSS + kk];  a2.y  = Al[r * LDSS + kk + 1];
      b2v.x = W3[n * HID + kk];   b2v.y = W3[n * HID + kk + 1];
      acc2  = wmma_f32(a2, b2v, acc2);   // interleaved chains hide WMMA RAW
    }
    const float bias  = b2[n] - b3[n];
    const int   mbase = m0 + ((lane >> 4) << 3);
#pragma unroll
    for (int j = 0; j < 8; ++j)
      X[(size_t)(mbase + j) * HID + n] = acc[j] - acc2[j] + bias;
  }
}

// ---------------------------------------------------------------------------
// Kernel 2: out = dual @ W1^T + b1 - sigma * rhs   (base for the scatter)
// ---------------------------------------------------------------------------
__global__ __launch_bounds__(128)
void dual_base_kernel(const float* __restrict__ dual,
                      const float* __restrict__ W1, const float* __restrict__ b1,
                      const float* __restrict__ rhs,
                      const float* __restrict__ sigma_p,
                      float* __restrict__ out) {
  __shared__ float Ad[16 * LDSS];
  const int m0  = blockIdx.x * 16;
  const int tid = threadIdx.x;

  const float4* gd = (const float4*)(dual + (size_t)m0 * HID);
#pragma unroll
  for (int j = 0; j < 4; ++j) {
    const int f   = tid + j * 128;
    const int row = f >> 5;
    const int c4  = f & 31;
    *(float4*)&Ad[row * LDSS + c4 * 4] = gd[f];
  }
  __syncthreads();

  const int   lane  = tid & 31;
  const int   wave  = tid >> 5;
  const int   r     = lane & 15;
  const int   koff  = (lane >> 4) << 1;
  const float sigma = sigma_p[0];

#pragma unroll
  for (int t = 0; t < 2; ++t) {
    const int n0 = (wave * 2 + t) * 16;
    const int n  = n0 + (lane & 15);
    v8f acc = {};
#pragma unroll
    for (int k0 = 0; k0 < HID; k0 += 4) {
      const int kk = k0 + koff;
      v2f a, b;
      a.x = Ad[r * LDSS + kk];  a.y = Ad[r * LDSS + kk + 1];
      b.x = W1[n * HID + kk];   b.y = W1[n * HID + kk + 1];
      acc = wmma_f32(a, b, acc);
    }
    const float bias  = b1[n];
    const int   mbase = m0 + ((lane >> 4) << 3);
#pragma unroll
    for (int j = 0; j < 8; ++j) {
      const size_t idx = (size_t)(mbase + j) * HID + n;
      out[idx] = acc[j] + bias - sigma * rhs[idx];
    }
  }
}

// ---------------------------------------------------------------------------
// Kernel 3: COO scatter — one wave per nonzero, 4 floats per lane.
// out[rows[e]] += sigma * vals[e] * X[cols[e]]; relaxed agent-scope f32
// atomics land in L2 (out is L2-resident).
// ---------------------------------------------------------------------------
__global__ __launch_bounds__(256)
void scatter_kernel(const int* __restrict__ rows, const int* __restrict__ cols,
                    const float* __restrict__ vals, const float* __restrict__ X,
                    const float* __restrict__ sigma_p, float* __restrict__ out,
                    int nnz) {
  const int e = blockIdx.x * 8 + (threadIdx.x >> 5);   // 8 waves per block
  if (e >= nnz) return;
  const int   lane = threadIdx.x & 31;
  const int   rr   = rows[e];
  const int   cc   = cols[e];
  const float s    = vals[e] * sigma_p[0];

  const float4 xv = ((const float4*)(X + (size_t)cc * HID))[lane];
  float* op = out + (size_t)rr * HID + lane * 4;
  __hip_atomic_fetch_add(&op[0], s * xv.x, __ATOMIC_RELAXED, __HIP_MEMORY_SCOPE_AGENT);
  __hip_atomic_fetch_add(&op[1], s * xv.y, __ATOMIC_RELAXED, __HIP_MEMORY_SCOPE_AGENT);
  __hip_atomic_fetch_add(&op[2], s * xv.z, __ATOMIC_RELAXED, __HIP_MEMORY_SCOPE_AGENT);
  __hip_atomic_fetch_add(&op[3], s * xv.w, __ATOMIC_RELAXED, __HIP_MEMORY_SCOPE_AGENT);
}

// ---------------------------------------------------------------------------
// Kernel 4: in-place leaky ReLU, float4 grid-stride.
// ---------------------------------------------------------------------------
__global__ __launch_bounds__(256)
void leaky_kernel(float* __restrict__ out, int n4) {
  int i = blockIdx.x * blockDim.x + threadIdx.x;
  const int stride = gridDim.x * blockDim.x;
  float4* p = (float4*)out;
  for (; i < n4; i += stride) {
    float4 v = p[i];
    v.x = v.x >= 0.f ? v.x : NEG_SLOPE * v.x;
    v.y = v.y >= 0.f ? v.y : NEG_SLOPE * v.y;
    v.z = v.z >= 0.f ? v.z : NEG_SLOPE * v.z;
    v.w = v.w >= 0.f ? v.w : NEG_SLOPE * v.w;
    p[i] = v;
  }
}

extern "C" void kernel_launch(void* const* d_in, const int* in_sizes, int n_in,
                              void* d_out, int out_size, void* d_ws, size_t ws_size,
                              hipStream_t stream) {
  const float* primal      = (const float*)d_in[0];
  const float* last_primal = (const float*)d_in[1];
  const float* dual        = (const float*)d_in[2];
  const int*   rows        = (const int*)d_in[3];
  const int*   cols        = (const int*)d_in[4];
  const float* vals        = (const float*)d_in[5];
  const float* rhs         = (const float*)d_in[6];
  const float* W1          = (const float*)d_in[7];
  const float* b1          = (const float*)d_in[8];
  const float* W2          = (const float*)d_in[9];
  const float* b2          = (const float*)d_in[10];
  const float* W3          = (const float*)d_in[11];
  const float* b3          = (const float*)d_in[12];
  const float* sigma       = (const float*)d_in[13];

  float* out = (float*)d_out;
  float* X   = (float*)d_ws;            // 200000*128*4 = 102.4 MB scratch

  const int n_vars = in_sizes[0] / HID; // 200000
  const int m_cons = in_sizes[2] / HID; // 100000
  const int nnz    = in_sizes[5];       // 1600000

  fused_x_kernel<<<n_vars / 16, 128, 0, stream>>>(primal, last_primal,
                                                  W2, b2, W3, b3, X);
  dual_base_kernel<<<m_cons / 16, 128, 0, stream>>>(dual, W1, b1, rhs, sigma, out);
  scatter_kernel<<<(nnz + 7) / 8, 256, 0, stream>>>(rows, cols, vals, X, sigma,
                                                    out, nnz);
  leaky_kernel<<<4096, 256, 0, stream>>>(out, (m_cons * HID) / 4);
}